// FeedForwardQuantum_60413009985889
// MI455X (gfx1250) — compile-verified
//
#include <hip/hip_runtime.h>
#include <hip/hip_bf16.h>

// ---------------------------------------------------------------------------
// FeedForwardQuantum for MI455X (gfx1250, wave32)
//   K1: ang = x(16384x1024) @ W1^T + b1        -> v_wmma_f32_16x16x4_f32
//   K2: 10-qubit statevector sim, 1 state per wave, state in 64 VGPRs,
//       lane exchanges via ds_swizzle_b32 (XOR mode)
//   K3: out = ev(16384x16) @ W2^T + b2          -> v_wmma_f32_16x16x4_f32
// ---------------------------------------------------------------------------

typedef float v2f __attribute__((ext_vector_type(2)));
typedef float v8f __attribute__((ext_vector_type(8)));

#define EMBED 1024
#define NROWS 16384      // B*S
#define NQ    10

// ---- lane exchange: ds_swizzle_b32, group-of-32 XOR mode ------------------
// offset = {1'b0, xor_mask[4:0], or_mask[4:0]=0, and_mask[4:0]=0x1f}
template <int XORMASK>
__device__ __forceinline__ float swz(float v) {
  int i = __builtin_bit_cast(int, v);
  i = __builtin_amdgcn_ds_swizzle(i, (XORMASK << 10) | 0x1f);
  return __builtin_bit_cast(float, i);
}

// ===========================================================================
// Kernel 1: angle GEMM.  One wave per 16-row tile, K=1024 in 4-wide steps.
// C layout (f32 16x16): vgpr i -> (M=i, N=lane) lanes0-15, (M=8+i, N=lane-16).
// A (16x4):  lane<16: M=lane, K={k0,k0+1}; lane>=16: M=lane-16, K={k0+2,k0+3}.
// B (4x16):  mirrored: lane<16: K rows {k0,k0+1} at N=lane; lanes>=16: {k0+2,k0+3}.
// N>=10 columns are masked (multiply-by-0, no predicated loads -> no exec
// branches, B pair merges into a single global_load_b64).
// ===========================================================================
__global__ void __launch_bounds__(128)
k_angles(const float* __restrict__ x, const float* __restrict__ W1,
         const float* __restrict__ b1, float* __restrict__ ang) {
  const int lane  = threadIdx.x & 31;
  const int tile  = blockIdx.x * 4 + (threadIdx.x >> 5);
  const int row0  = tile * 16;
  const int l15   = lane & 15;
  const int khalf = (lane >> 4) << 1;                 // 0 or 2

  const float* __restrict__ xrow = x + (size_t)(row0 + l15) * EMBED;
  const bool  nvalid = l15 < NQ;
  const float msk = nvalid ? 1.0f : 0.0f;
  // clamp to row 0 so loads are always in-bounds; result masked to 0
  const float* __restrict__ wrow = W1 + (size_t)(nvalid ? l15 : 0) * EMBED;

  const float bias = nvalid ? b1[l15] : 0.0f;         // bias along N
  v8f c;
#pragma unroll
  for (int i = 0; i < 8; ++i) c[i] = bias;

#pragma unroll 8
  for (int k0 = 0; k0 < EMBED; k0 += 4) {
    v2f a, b;
    a.x = xrow[k0 + khalf];
    a.y = xrow[k0 + khalf + 1];
    b.x = wrow[k0 + khalf]     * msk;
    b.y = wrow[k0 + khalf + 1] * msk;
    c = __builtin_amdgcn_wmma_f32_16x16x4_f32(false, a, false, b,
                                              (short)0, c, false, false);
  }

  const int mbase = row0 + ((lane >> 4) << 3);
#pragma unroll
  for (int i = 0; i < 8; ++i)
    ang[(size_t)(mbase + i) * 16 + l15] = c[i];       // cols 10..15 == 0
}

// ===========================================================================
// Kernel 2: quantum simulator. amp index a = hi*32 + lane (hi = a[9:5]).
// Qubit q stride = 2^(9-q): q0..q4 -> hi bits {16,8,4,2,1},
//                           q5..q9 -> lane bits {16,8,4,2,1}.
// ===========================================================================

// RX on hi-bit qubit: uniform update new = c*mine + s*(partner.im, -partner.re)
template <int HM>
__device__ __forceinline__ void rx_hi(float* re, float* im, float c, float s) {
#pragma unroll
  for (int h = 0; h < 32; ++h) {
    if (h & HM) continue;
    const int h1 = h | HM;
    float r0 = re[h], i0 = im[h], r1 = re[h1], i1 = im[h1];
    re[h]  = c * r0 + s * i1;
    im[h]  = c * i0 - s * r1;
    re[h1] = c * r1 + s * i0;
    im[h1] = c * i1 - s * r0;
  }
}

// RX on lane-bit qubit (partner via swizzle; update rule is bit-symmetric)
template <int LM>
__device__ __forceinline__ void rx_ln(float* re, float* im, float c, float s) {
#pragma unroll
  for (int h = 0; h < 32; ++h) {
    float pr = swz<LM>(re[h]);
    float pi = swz<LM>(im[h]);
    float r = re[h], i = im[h];
    re[h] = c * r + s * pi;
    im[h] = c * i - s * pr;
  }
}

// CNOT: control hi-bit, target hi-bit  -> static register swaps
template <int CM, int TM>
__device__ __forceinline__ void cnot_hh(float* re, float* im) {
#pragma unroll
  for (int h = 0; h < 32; ++h) {
    if (!(h & CM) || (h & TM)) continue;
    const int h1 = h | TM;
    float t = re[h]; re[h] = re[h1]; re[h1] = t;
    t = im[h]; im[h] = im[h1]; im[h1] = t;
  }
}

// CNOT: control hi-bit, target lane-bit -> full swizzle of the controlled slice
template <int CM, int LTM>
__device__ __forceinline__ void cnot_hl(float* re, float* im) {
#pragma unroll
  for (int h = 0; h < 32; ++h) {
    if (!(h & CM)) continue;
    re[h] = swz<LTM>(re[h]);
    im[h] = swz<LTM>(im[h]);
  }
}

// CNOT: control lane-bit, target lane-bit -> swizzle + per-lane select
template <int CLM, int TLM>
__device__ __forceinline__ void cnot_ll(int lane, float* re, float* im) {
  const bool ctl = (lane & CLM) != 0;
#pragma unroll
  for (int h = 0; h < 32; ++h) {
    float sr = swz<TLM>(re[h]);
    float si = swz<TLM>(im[h]);
    re[h] = ctl ? sr : re[h];
    im[h] = ctl ? si : im[h];
  }
}

// CNOT: control lane-bit, target hi-bit -> per-lane conditional register swap
template <int CLM, int THM>
__device__ __forceinline__ void cnot_lh(int lane, float* re, float* im) {
  const bool ctl = (lane & CLM) != 0;
#pragma unroll
  for (int h = 0; h < 32; ++h) {
    if (h & THM) continue;
    const int h1 = h | THM;
    float r0 = re[h], r1 = re[h1];
    re[h]  = ctl ? r1 : r0;
    re[h1] = ctl ? r0 : r1;
    float i0 = im[h], i1 = im[h1];
    im[h]  = ctl ? i1 : i0;
    im[h1] = ctl ? i0 : i1;
  }
}

__device__ __forceinline__ void cnot_ring(int lane, float* re, float* im) {
  cnot_hh<16, 8>(re, im);          // q0 -> q1
  cnot_hh<8, 4>(re, im);           // q1 -> q2
  cnot_hh<4, 2>(re, im);           // q2 -> q3
  cnot_hh<2, 1>(re, im);           // q3 -> q4
  cnot_hl<1, 16>(re, im);          // q4 -> q5
  cnot_ll<16, 8>(lane, re, im);    // q5 -> q6
  cnot_ll<8, 4>(lane, re, im);     // q6 -> q7
  cnot_ll<4, 2>(lane, re, im);     // q7 -> q8
  cnot_ll<2, 1>(lane, re, im);     // q8 -> q9
  cnot_lh<1, 16>(lane, re, im);    // q9 -> q0
}

__global__ void __launch_bounds__(128)
k_qsim(const float* __restrict__ qw, float* __restrict__ buf) {
  const int lane  = threadIdx.x & 31;
  const int state = blockIdx.x * 4 + (threadIdx.x >> 5);
  float* __restrict__ srow = buf + (size_t)state * 16;

  // merged angles: data RX + trainable layer0 RX fuse (RX(a)RX(b)=RX(a+b))
  float c0[NQ], s0[NQ], c1[NQ], s1[NQ];
#pragma unroll
  for (int q = 0; q < NQ; ++q) {
    const float t0 = (srow[q] + qw[q]) * 0.5f;
    c0[q] = __cosf(t0); s0[q] = __sinf(t0);
    const float t1 = qw[NQ + q] * 0.5f;
    c1[q] = __cosf(t1); s1[q] = __sinf(t1);
  }

  // closed-form product state: amp(a) = Prod(bit? sin : cos) * (-i)^popc(a)
  float re[32], im[32];
  float plane = 1.0f;
  plane *= (lane & 16) ? s0[5] : c0[5];
  plane *= (lane & 8)  ? s0[6] : c0[6];
  plane *= (lane & 4)  ? s0[7] : c0[7];
  plane *= (lane & 2)  ? s0[8] : c0[8];
  plane *= (lane & 1)  ? s0[9] : c0[9];
  const int kl = __popc(lane & 31);
#pragma unroll
  for (int h = 0; h < 32; ++h) {
    float p = plane;
    p *= (h & 16) ? s0[0] : c0[0];
    p *= (h & 8)  ? s0[1] : c0[1];
    p *= (h & 4)  ? s0[2] : c0[2];
    p *= (h & 2)  ? s0[3] : c0[3];
    p *= (h & 1)  ? s0[4] : c0[4];
    const int k = (kl + __popc(h)) & 3;            // (-i)^k
    re[h] = (k == 0) ? p : ((k == 2) ? -p : 0.0f);
    im[h] = (k == 1) ? -p : ((k == 3) ? p : 0.0f);
  }

  cnot_ring(lane, re, im);                          // layer 0 entangler

  rx_hi<16>(re, im, c1[0], s1[0]);                  // layer 1 RX
  rx_hi<8>(re, im, c1[1], s1[1]);
  rx_hi<4>(re, im, c1[2], s1[2]);
  rx_hi<2>(re, im, c1[3], s1[3]);
  rx_hi<1>(re, im, c1[4], s1[4]);
  rx_ln<16>(re, im, c1[5], s1[5]);
  rx_ln<8>(re, im, c1[6], s1[6]);
  rx_ln<4>(re, im, c1[7], s1[7]);
  rx_ln<2>(re, im, c1[8], s1[8]);
  rx_ln<1>(re, im, c1[9], s1[9]);

  cnot_ring(lane, re, im);                          // layer 1 entangler

  // <Z_q>: signed probability sums
  float tot = 0.f, e0 = 0.f, e1 = 0.f, e2 = 0.f, e3 = 0.f, e4 = 0.f;
#pragma unroll
  for (int h = 0; h < 32; ++h) {
    const float p = re[h] * re[h] + im[h] * im[h];
    tot += p;
    e0 += (h & 16) ? -p : p;
    e1 += (h & 8)  ? -p : p;
    e2 += (h & 4)  ? -p : p;
    e3 += (h & 2)  ? -p : p;
    e4 += (h & 1)  ? -p : p;
  }
  float ev[NQ] = {
    e0, e1, e2, e3, e4,
    (lane & 16) ? -tot : tot, (lane & 8) ? -tot : tot,
    (lane & 4)  ? -tot : tot, (lane & 2) ? -tot : tot,
    (lane & 1)  ? -tot : tot };

#pragma unroll
  for (int q = 0; q < NQ; ++q) {                    // wave32 butterfly reduce
    float v = ev[q];
    v += swz<16>(v); v += swz<8>(v); v += swz<4>(v); v += swz<2>(v); v += swz<1>(v);
    ev[q] = v;
  }
  if (lane == 0) {
#pragma unroll
    for (int q = 0; q < NQ; ++q) srow[q] = ev[q];   // cols 10..15 stay 0
  }
}

// ===========================================================================
// Kernel 3: output GEMM. One wave per 16x16 tile of out; K=12 (3 WMMAs).
// ev cols 10,11 are zero by construction, so B rows k>=10 only need to be
// FINITE, not zero: clamp the index (always in-bounds) and mask.
// ===========================================================================
__global__ void __launch_bounds__(128)
k_out(const float* __restrict__ ev, const float* __restrict__ W2,
      const float* __restrict__ b2, float* __restrict__ out) {
  const int lane  = threadIdx.x & 31;
  const int tile  = blockIdx.x * 4 + (threadIdx.x >> 5);
  const int mt    = tile >> 6;                      // 0..1023
  const int nt    = tile & 63;                      // 0..63
  const int row0  = mt * 16, col0 = nt * 16;
  const int l15   = lane & 15;
  const int khalf = (lane >> 4) << 1;

  const float bias = b2[col0 + l15];
  v8f c;
#pragma unroll
  for (int i = 0; i < 8; ++i) c[i] = bias;

  const float* __restrict__ arow = ev + (size_t)(row0 + l15) * 16;
  const float* __restrict__ bcol = W2 + (size_t)(col0 + l15) * NQ;  // W2[e][q]

#pragma unroll
  for (int k0 = 0; k0 < 12; k0 += 4) {
    const int k   = k0 + khalf;
    const int kb0 = (k     < NQ) ? k     : 0;       // clamped, in-bounds
    const int kb1 = (k + 1 < NQ) ? k + 1 : 0;
    const float m0 = (k     < NQ) ? 1.0f : 0.0f;
    const float m1 = (k + 1 < NQ) ? 1.0f : 0.0f;
    v2f a, b;
    a.x = arow[k];
    a.y = arow[k + 1];
    b.x = bcol[kb0] * m0;
    b.y = bcol[kb1] * m1;
    c = __builtin_amdgcn_wmma_f32_16x16x4_f32(false, a, false, b,
                                              (short)0, c, false, false);
  }

  const int mbase = row0 + ((lane >> 4) << 3);
#pragma unroll
  for (int i = 0; i < 8; ++i)
    out[(size_t)(mbase + i) * EMBED + col0 + l15] = c[i];
}

// ===========================================================================
extern "C" void kernel_launch(void* const* d_in, const int* in_sizes, int n_in,
                              void* d_out, int out_size, void* d_ws, size_t ws_size,
                              hipStream_t stream) {
  const float* x  = (const float*)d_in[0];   // (16,1024,1024)
  const float* W1 = (const float*)d_in[1];   // (10,1024)
  const float* b1 = (const float*)d_in[2];   // (10)
  const float* qw = (const float*)d_in[3];   // (2,10)
  const float* W2 = (const float*)d_in[4];   // (1024,10)
  const float* b2 = (const float*)d_in[5];   // (1024)
  float* out = (float*)d_out;                // (16,1024,1024)
  float* buf = (float*)d_ws;                 // 16384 x 16 f32 (1 MB): ang -> ev

  k_angles<<<NROWS / 16 / 4, 128, 0, stream>>>(x, W1, b1, buf);
  k_qsim  <<<NROWS / 4,      128, 0, stream>>>(qw, buf);
  k_out   <<<NROWS / 16 * 64 / 4, 128, 0, stream>>>(buf, W2, b2, out);
}